// Unpooling_45578192945215
// MI455X (gfx1250) — compile-verified
//
#include <hip/hip_runtime.h>
#include <math.h>

typedef __attribute__((ext_vector_type(2))) float v2f;
typedef __attribute__((ext_vector_type(8))) float v8f;

#define KNN_K      3
#define CX         128
#define CIN        256   // Cx + Cy
#define ODIM       256
#define MAXNX      4096
#define FPAD       260   // 256 + 4 pad: row stride in LDS dwords (260 % 64 == 4 -> conflict-free column reads)
#define LN_EPS     1e-5f
#define W_EPS      1e-16f

// ---------------------------------------------------------------------------
// Kernel A: per-query top-3 nearest neighbors within the same batch.
// pos_x / batch_x staged in LDS (SoA); sorted batches -> binary search range.
// Emits normalized inverse-d^2 weights and neighbor indices to workspace.
// ---------------------------------------------------------------------------
__global__ __launch_bounds__(256) void knn_top3_kernel(
    const float* __restrict__ pos_x, const int* __restrict__ batch_x,
    const float* __restrict__ pos_y, const int* __restrict__ batch_y,
    int Nx, int Ny, int* __restrict__ out_idx, float* __restrict__ out_w)
{
    __shared__ float spx[MAXNX];
    __shared__ float spy[MAXNX];
    __shared__ float spz[MAXNX];
    __shared__ int   sbx[MAXNX];

    const int tid = threadIdx.x;
    for (int j = tid; j < Nx; j += 256) {
        spx[j] = pos_x[3 * j + 0];
        spy[j] = pos_x[3 * j + 1];
        spz[j] = pos_x[3 * j + 2];
        sbx[j] = batch_x[j];
    }
    __syncthreads();

    const int n = blockIdx.x * 256 + tid;
    if (n >= Ny) return;

    const float qx = pos_y[3 * n + 0];
    const float qy = pos_y[3 * n + 1];
    const float qz = pos_y[3 * n + 2];
    const int   b  = batch_y[n];

    // lower_bound(b), lower_bound(b+1) over sorted batch_x in LDS
    int lo, hi;
    { int l = 0, r = Nx; while (l < r) { int m = (l + r) >> 1; if (sbx[m] < b) l = m + 1; else r = m; } lo = l; }
    { int l = lo, r = Nx; while (l < r) { int m = (l + r) >> 1; if (sbx[m] < b + 1) l = m + 1; else r = m; } hi = l; }

    float d0 = 3.0e38f, d1 = 3.0e38f, d2 = 3.0e38f;
    int   i0 = lo, i1 = lo, i2 = lo;
    for (int j = lo; j < hi; ++j) {
        const float dx = qx - spx[j];
        const float dy = qy - spy[j];
        const float dz = qz - spz[j];
        const float dd = dx * dx + dy * dy + dz * dz;
        if (dd < d2) {
            if (dd < d0)      { d2 = d1; i2 = i1; d1 = d0; i1 = i0; d0 = dd; i0 = j; }
            else if (dd < d1) { d2 = d1; i2 = i1; d1 = dd; i1 = j; }
            else              { d2 = dd; i2 = j; }
        }
    }

    const float w0 = 1.0f / fmaxf(d0, W_EPS);
    const float w1 = 1.0f / fmaxf(d1, W_EPS);
    const float w2 = 1.0f / fmaxf(d2, W_EPS);
    const float inv = 1.0f / (w0 + w1 + w2);

    out_idx[3 * n + 0] = i0;
    out_idx[3 * n + 1] = i1;
    out_idx[3 * n + 2] = i2;
    out_w[3 * n + 0] = w0 * inv;
    out_w[3 * n + 1] = w1 * inv;
    out_w[3 * n + 2] = w2 * inv;
}

// ---------------------------------------------------------------------------
// Kernel B: per 16-row tile
//   1) gather weighted x-features + concat y into LDS (16 x 256, padded rows)
//   2) GEMM with V_WMMA_F32_16X16X4_F32: each wave owns 2 N-tiles, 64 k-steps
//   3) LayerNorm (+ ReLU) via wave32 shfl_xor reductions
// ---------------------------------------------------------------------------
__global__ __launch_bounds__(256) void gemm_ln_relu_kernel(
    const float* __restrict__ x, const float* __restrict__ y,
    const float* __restrict__ W, const float* __restrict__ gamma,
    const float* __restrict__ beta, const int* __restrict__ nn_idx,
    const float* __restrict__ nn_w, float* __restrict__ out)
{
    __shared__ float feats[16 * FPAD];
    __shared__ float hbuf[16 * FPAD];
    __shared__ int   sidx[16 * KNN_K];
    __shared__ float sw[16 * KNN_K];

    const int tid  = threadIdx.x;
    const int lane = tid & 31;
    const int wv   = tid >> 5;
    const int m0   = blockIdx.x * 16;

    if (tid < 16 * KNN_K) {
        sidx[tid] = nn_idx[m0 * KNN_K + tid];
        sw[tid]   = nn_w[m0 * KNN_K + tid];
    }
    __syncthreads();

    // Build the 16 x 256 concat(interp, y) tile in LDS.
    // tid stepping by 256 keeps j = tid & 255 fixed per thread (column owner).
    for (int e = tid; e < 16 * CIN; e += 256) {
        const int m = e >> 8;
        const int j = e & 255;
        float v;
        if (j < CX) {
            const int ib = m * KNN_K;
            v = sw[ib + 0] * x[sidx[ib + 0] * CX + j]
              + sw[ib + 1] * x[sidx[ib + 1] * CX + j]
              + sw[ib + 2] * x[sidx[ib + 2] * CX + j];
        } else {
            v = y[(m0 + m) * (CIN - CX) + (j - CX)];
        }
        feats[m * FPAD + j] = v;
    }
    __syncthreads();

    // f32 WMMA 16x16x4: A 16x4 (2 VGPR), B 4x16 (2 VGPR), C/D 16x16 (8 VGPR).
    // A lane layout: lanes 0-15 -> M=lane, K = k0+{0,1}; lanes 16-31 -> K = k0+{2,3}.
    // B lane layout: lanes 0-15 -> N=lane, K = k0+{0,1}; lanes 16-31 -> K = k0+{2,3}.
    const int mrow  = lane & 15;
    const int khalf = (lane >> 4) << 1;

    for (int nt = wv; nt < ODIM / 16; nt += 8) {
        v8f acc = {};
        const int ncol = nt * 16 + (lane & 15);
        const float* wrow = W + (size_t)ncol * CIN;
        for (int k0 = 0; k0 < CIN; k0 += 4) {
            const v2f a = *(const v2f*)&feats[mrow * FPAD + k0 + khalf];
            v2f bvec;
            bvec.x = wrow[k0 + khalf + 0];   // B[k][n] = W[n][k]  (h = feats @ W^T)
            bvec.y = wrow[k0 + khalf + 1];
            acc = __builtin_amdgcn_wmma_f32_16x16x4_f32(
                false, a, false, bvec, (short)0, acc, false, false);
        }
        // C/D layout: VGPR r -> row r (lanes 0-15) / row r+8 (lanes 16-31), N = lane & 15.
#pragma unroll
        for (int r = 0; r < 8; ++r) {
            const int mm = (lane < 16) ? r : (r + 8);
            hbuf[mm * FPAD + nt * 16 + (lane & 15)] = acc[r];
        }
    }
    __syncthreads();

    // LayerNorm + ReLU: wave wv handles rows 2*wv and 2*wv+1 (8 waves x 2 = 16 rows).
    for (int row = wv * 2; row < wv * 2 + 2; ++row) {
        float vals[ODIM / 32];
        float s = 0.0f, s2 = 0.0f;
#pragma unroll
        for (int i = 0; i < ODIM / 32; ++i) {
            const float v = hbuf[row * FPAD + lane + i * 32];
            vals[i] = v;
            s += v;
            s2 += v * v;
        }
#pragma unroll
        for (int off = 16; off >= 1; off >>= 1) {
            s  += __shfl_xor(s, off, 32);
            s2 += __shfl_xor(s2, off, 32);
        }
        const float mu  = s * (1.0f / ODIM);
        const float var = s2 * (1.0f / ODIM) - mu * mu;   // jnp.var form
        const float rs  = rsqrtf(var + LN_EPS);
        float* orow = out + (size_t)(m0 + row) * ODIM;
#pragma unroll
        for (int i = 0; i < ODIM / 32; ++i) {
            const int c = lane + i * 32;
            const float hv = (vals[i] - mu) * rs * gamma[c] + beta[c];
            orow[c] = fmaxf(hv, 0.0f);
        }
    }
}

extern "C" void kernel_launch(void* const* d_in, const int* in_sizes, int n_in,
                              void* d_out, int out_size, void* d_ws, size_t ws_size,
                              hipStream_t stream) {
    const float* pos_x   = (const float*)d_in[0];
    const float* x       = (const float*)d_in[1];
    const int*   batch_x = (const int*)d_in[2];
    const float* pos_y   = (const float*)d_in[3];
    const float* y       = (const float*)d_in[4];
    const int*   batch_y = (const int*)d_in[5];
    const float* W       = (const float*)d_in[6];
    const float* gamma   = (const float*)d_in[7];
    const float* beta    = (const float*)d_in[8];
    float* out = (float*)d_out;

    const int Nx = in_sizes[2];
    const int Ny = in_sizes[5];

    int*   ws_idx = (int*)d_ws;
    float* ws_w   = (float*)((char*)d_ws + (size_t)Ny * KNN_K * sizeof(int));

    knn_top3_kernel<<<(Ny + 255) / 256, 256, 0, stream>>>(
        pos_x, batch_x, pos_y, batch_y, Nx, Ny, ws_idx, ws_w);

    gemm_ln_relu_kernel<<<Ny / 16, 256, 0, stream>>>(
        x, y, W, gamma, beta, ws_idx, ws_w, out);
}